// PCENetwork_36730560315884
// MI455X (gfx1250) — compile-verified
//
#include <hip/hip_runtime.h>
#include <hip/hip_bf16.h>
#include <math.h>

// ---------------------------------------------------------------------------
// MoE patch network for MI455X (gfx1250, wave32).
// Key simplifications vs reference: routing buffer is identity on valid tokens;
// img is never materialized (pooling fused); classifier GEMM uses
// V_WMMA_F32_16X16X4_F32 (M=16 output rows == WMMA tile height).
// All reductions deterministic (no FP atomics).
// ---------------------------------------------------------------------------

#define NTOK   4096      // 16 images * 256 patches
#define CCAP   640       // ceil(4096*1.25/8)
#define NEXP   8
#define NCLS   1000
#define KDIM   512

typedef float v2f __attribute__((ext_vector_type(2)));
typedef float v8f __attribute__((ext_vector_type(8)));

// ---------------- workspace layout (bytes) ----------------
#define WS_EIDX    0                        // 4096 int
#define WS_GATE    (16*1024)                // 4096 float (top prob, later masked gate)
#define WS_PROBS   (32*1024)                // 4096*8 float
#define WS_TOKSUM  (WS_PROBS + 128*1024)    // 4096*8 float
#define WS_POOLED  (WS_TOKSUM + 128*1024)   // 16*512 float  (== feat)

// ===========================================================================
// Kernel 1: gating.  One thread per token.
// logits[e] = gate_b[e] + sum_c gate_w[e,c] * mean_{hw}(coords[n,c])
// ===========================================================================
__global__ __launch_bounds__(256) void k1_gate(
    const float* __restrict__ X,       // (16,3,256,256)
    const float* __restrict__ gw,      // (8,23)
    const float* __restrict__ gb,      // (8)
    int*   __restrict__ e_idx,         // (4096)
    float* __restrict__ gate_p,        // (4096)
    float* __restrict__ probs)         // (4096,8)
{
    const int n = blockIdx.x * 256 + threadIdx.x;
    const int b = n >> 8, p = n & 255, r = p >> 4, c = p & 15;

    // --- patch means of the 3 image channels (16x16 each, float4 loads) ---
    float mx[3];
    const float* xb = X + (size_t)b * 3 * 65536;
    #pragma unroll
    for (int ch = 0; ch < 3; ++ch) {
        float s = 0.f;
        const float* base = xb + ch * 65536 + (r * 16) * 256 + c * 16;
        #pragma unroll
        for (int i = 0; i < 16; ++i) {
            const float4* row = (const float4*)(base + i * 256);
            float4 a0 = row[0], a1 = row[1], a2 = row[2], a3 = row[3];
            s += a0.x + a0.y + a0.z + a0.w + a1.x + a1.y + a1.z + a1.w
               + a2.x + a2.y + a2.z + a2.w + a3.x + a3.y + a3.z + a3.w;
        }
        mx[ch] = s * (1.f / 256.f);
    }

    // --- analytic means of the 20 positional channels over the patch ---
    float pm[20];
    const float fc = (float)c, fr = (float)r;
    pm[0] = (fc * 16.f + 7.5f) * (1.f / 256.f);   // mean gx
    pm[1] = (fr * 16.f + 7.5f) * (1.f / 256.f);   // mean gy
    pm[2] = (fc + 0.5f) * (1.f / 16.f);           // pcx
    pm[3] = (fr + 0.5f) * (1.f / 16.f);           // pcy
    int idx = 4;
    #pragma unroll
    for (int f = 0; f < 2; ++f) {
        const float s = 6.28318530717958647692f * (float)(1 << f);
        float sgx = 0.f, cgx = 0.f, sgy = 0.f, cgy = 0.f;
        #pragma unroll
        for (int j = 0; j < 16; ++j) {
            float tx = s * (fc * 16.f + (float)j) * (1.f / 256.f);
            float ty = s * (fr * 16.f + (float)j) * (1.f / 256.f);
            sgx += __sinf(tx); cgx += __cosf(tx);
            sgy += __sinf(ty); cgy += __cosf(ty);
        }
        pm[idx++] = sgx * (1.f / 16.f);
        pm[idx++] = cgx * (1.f / 16.f);
        pm[idx++] = sgy * (1.f / 16.f);
        pm[idx++] = cgy * (1.f / 16.f);
        pm[idx++] = __sinf(s * pm[2]);
        pm[idx++] = __cosf(s * pm[2]);
        pm[idx++] = __sinf(s * pm[3]);
        pm[idx++] = __cosf(s * pm[3]);
    }

    // --- logits, softmax, argmax ---
    float lg[NEXP], mmax = -1e30f;
    #pragma unroll
    for (int e = 0; e < NEXP; ++e) {
        const float* w = gw + e * 23;
        float v = gb[e] + w[0] * mx[0] + w[1] * mx[1] + w[2] * mx[2];
        #pragma unroll
        for (int k = 0; k < 20; ++k) v += w[3 + k] * pm[k];
        lg[e] = v;
        mmax = fmaxf(mmax, v);
    }
    float se = 0.f, pe[NEXP];
    #pragma unroll
    for (int e = 0; e < NEXP; ++e) { pe[e] = __expf(lg[e] - mmax); se += pe[e]; }
    const float inv = 1.f / se;
    int best = 0; float bl = lg[0];
    #pragma unroll
    for (int e = 1; e < NEXP; ++e) if (lg[e] > bl) { bl = lg[e]; best = e; }

    e_idx[n]  = best;
    gate_p[n] = pe[best] * inv;
    #pragma unroll
    for (int e = 0; e < NEXP; ++e) probs[n * NEXP + e] = pe[e] * inv;
}

// ===========================================================================
// Kernel 2: ordered capacity scan + aux loss.  Single block of 256 threads,
// 16 chunks in token order; per-chunk rank via LDS; int shared atomics only.
// Overwrites gate_p with the capacity-masked gate.  Writes aux to out[16000].
// ===========================================================================
__global__ __launch_bounds__(256) void k2_capacity(
    const int*  __restrict__ e_idx,
    float*      __restrict__ gate_p,     // inout -> masked gates
    const float* __restrict__ probs,
    float*      __restrict__ out)        // d_out (aux at [16000])
{
    __shared__ int   sh_e[256];
    __shared__ int   sh_hist[NEXP];
    __shared__ int   base[NEXP];
    __shared__ int   vcnt[NEXP];
    __shared__ float red[256];
    __shared__ float ptot[NEXP];

    const int tid = threadIdx.x;
    if (tid < NEXP) { base[tid] = 0; vcnt[tid] = 0; }
    float psum[NEXP];
    #pragma unroll
    for (int e = 0; e < NEXP; ++e) psum[e] = 0.f;
    __syncthreads();

    for (int chunk = 0; chunk < 16; ++chunk) {
        const int n = chunk * 256 + tid;
        const int e = e_idx[n];
        sh_e[tid] = e;
        if (tid < NEXP) sh_hist[tid] = 0;
        __syncthreads();

        int rank = 0;
        for (int t = 0; t < tid; ++t) rank += (sh_e[t] == e) ? 1 : 0;
        const int  slot  = base[e] + rank;
        const bool valid = slot < CCAP;
        gate_p[n] = valid ? gate_p[n] : 0.f;
        if (valid) atomicAdd(&vcnt[e], 1);
        atomicAdd(&sh_hist[e], 1);

        #pragma unroll
        for (int e2 = 0; e2 < NEXP; ++e2) psum[e2] += probs[n * NEXP + e2];

        __syncthreads();
        if (tid < NEXP) base[tid] += sh_hist[tid];
        __syncthreads();
    }

    // deterministic tree-reduce of psum over the 256 threads
    for (int e = 0; e < NEXP; ++e) {
        red[tid] = psum[e];
        __syncthreads();
        for (int s = 128; s > 0; s >>= 1) {
            if (tid < s) red[tid] += red[tid + s];
            __syncthreads();
        }
        if (tid == 0) ptot[e] = red[0];
        __syncthreads();
    }

    if (tid == 0) {
        float aux = 0.f;
        #pragma unroll
        for (int e = 0; e < NEXP; ++e) {
            float fe = (float)vcnt[e] * (1.f / (float)NTOK);
            float pmean = ptot[e] * (1.f / (float)NTOK);
            aux += fe * pmean;
        }
        out[16000] = (float)NEXP * aux;
    }
}

// ===========================================================================
// Kernel 3: per-token expert conv (3x3, 3->8 ch, zero pad) + pixel sum.
// One block per token, one thread per pixel.  tokSum[n][co] =
//   gate[n] * sum_{i,j} relu(conv(Xp[n])[co,i,j] + b[co])      (0 if dropped)
// ===========================================================================
__global__ __launch_bounds__(256) void k3_conv(
    const float* __restrict__ X,         // (16,3,256,256)
    const float* __restrict__ ew,        // (8,8,3,3,3)
    const float* __restrict__ eb,        // (8,8)
    const int*   __restrict__ e_idx,
    const float* __restrict__ gates,
    float*       __restrict__ tokSum)    // (4096,8)
{
    const int n   = blockIdx.x;
    const int tid = threadIdx.x;

    __shared__ float sgate;
    __shared__ float s_x[3][18][18];     // haloed patch
    __shared__ float s_w[NEXP * 27];
    __shared__ float s_b[NEXP];
    __shared__ float s_red[256];

    if (tid == 0) sgate = gates[n];
    __syncthreads();

    if (sgate == 0.f) {                  // dropped (or zero-gate) token
        if (tid < NEXP) tokSum[n * NEXP + tid] = 0.f;
        return;
    }

    const int b = n >> 8, p = n & 255, r = p >> 4, c = p & 15;
    const int e = e_idx[n];

    // zero halo, then load 3x16x16 interior
    for (int lin = tid; lin < 3 * 18 * 18; lin += 256)
        ((float*)s_x)[lin] = 0.f;
    __syncthreads();
    for (int lin = tid; lin < 3 * 256; lin += 256) {
        const int ch = lin >> 8, pix = lin & 255, ii = pix >> 4, jj = pix & 15;
        s_x[ch][ii + 1][jj + 1] =
            X[(size_t)b * 3 * 65536 + ch * 65536 + (r * 16 + ii) * 256 + (c * 16 + jj)];
    }
    if (tid < NEXP * 27) s_w[tid] = ew[e * NEXP * 27 + tid];
    if (tid < NEXP)      s_b[tid] = eb[e * NEXP + tid];
    __syncthreads();

    const int i = tid >> 4, j = tid & 15;
    float acc[NEXP];
    #pragma unroll
    for (int co = 0; co < NEXP; ++co) {
        float v = s_b[co];
        #pragma unroll
        for (int ci = 0; ci < 3; ++ci)
            #pragma unroll
            for (int di = 0; di < 3; ++di)
                #pragma unroll
                for (int dj = 0; dj < 3; ++dj)
                    v += s_x[ci][i + di][j + dj] * s_w[co * 27 + ci * 9 + di * 3 + dj];
        acc[co] = fmaxf(v, 0.f);
    }

    // per-channel deterministic block reduction of the 256 pixel values
    #pragma unroll
    for (int co = 0; co < NEXP; ++co) {
        s_red[tid] = acc[co];
        __syncthreads();
        for (int s = 128; s > 0; s >>= 1) {
            if (tid < s) s_red[tid] += s_red[tid + s];
            __syncthreads();
        }
        if (tid == 0) tokSum[n * NEXP + co] = sgate * s_red[0];
        __syncthreads();
    }
}

// ===========================================================================
// Kernel 4: 32x32 average pool, fused across the 4 contributing patches.
// feat[b, co*64 + py*8 + px] = (1/1024) * sum of the 4 tokens' pixel sums.
// ===========================================================================
__global__ __launch_bounds__(256) void k4_pool(
    const float* __restrict__ tokSum,    // (4096,8)
    float*       __restrict__ feat)      // (16,512)
{
    const int cell = blockIdx.x * 256 + threadIdx.x;   // 8192 cells
    const int b  = cell >> 9;
    const int co = (cell >> 6) & 7;
    const int py = (cell >> 3) & 7;
    const int px = cell & 7;
    float s = 0.f;
    #pragma unroll
    for (int dr = 0; dr < 2; ++dr)
        #pragma unroll
        for (int dc = 0; dc < 2; ++dc) {
            const int n = b * 256 + (2 * py + dr) * 16 + (2 * px + dc);
            s += tokSum[n * NEXP + co];
        }
    feat[cell] = s * (1.f / 1024.f);
}

// ===========================================================================
// Kernel 5: classifier GEMM with V_WMMA_F32_16X16X4_F32.
// C(16 x 1000) = feat(16 x 512) @ lin_w(512 x 1000) + lin_b.
// One wave per 16-column tile; K swept in 128 steps of 4.
// A layout (16x4 f32): lanes 0-15 hold (M=lane, K=k0/k0+1), lanes 16-31 hold
// (M=lane-16, K=k0+2/k0+3).  B layout (4x16): VGPR0 rows K=k0 (lo lanes) /
// K=k0+2 (hi lanes), VGPR1 rows K=k0+1 / K=k0+3.
// Out-of-range columns are CLAMPED for loads (no divergent exec masking in the
// inner loop — keeps it a pure load+wmma pipeline) and masked only at store.
// ===========================================================================
__global__ __launch_bounds__(32) void k5_gemm_wmma(
    const float* __restrict__ feat,      // (16,512)
    const float* __restrict__ lw,        // (512,1000)
    const float* __restrict__ lb,        // (1000)
    float*       __restrict__ out)       // (16,1000) at d_out[0..15999]
{
    const int lane = threadIdx.x;
    const int n0   = blockIdx.x * 16;
    const int mrow = lane & 15;
    const int kk   = (lane < 16) ? 0 : 2;
    const int colr = n0 + mrow;                       // real column
    const int col  = (colr < NCLS) ? colr : (NCLS - 1);  // clamped for loads

    const float* ap = feat + mrow * KDIM + kk;        // A: two consecutive K
    const float* bp = lw + (size_t)kk * NCLS + col;   // B: two rows, stride NCLS

    v8f acc = {};
    for (int k0 = 0; k0 < KDIM; k0 += 4) {
        v2f a, bb;
        a.x  = ap[k0];
        a.y  = ap[k0 + 1];
        bb.x = bp[(size_t)k0 * NCLS];
        bb.y = bp[(size_t)(k0 + 1) * NCLS];
        acc = __builtin_amdgcn_wmma_f32_16x16x4_f32(
                  /*neg_a=*/false, a, /*neg_b=*/false, bb,
                  /*c_mod=*/(short)0, acc, /*reuse_a=*/false, /*reuse_b=*/false);
    }

    if (colr < NCLS) {
        const float bias = lb[colr];
        const int mo = (lane < 16) ? 0 : 8;
        #pragma unroll
        for (int g = 0; g < 8; ++g)
            out[(g + mo) * NCLS + colr] = acc[g] + bias;
    }
}

// ===========================================================================
extern "C" void kernel_launch(void* const* d_in, const int* in_sizes, int n_in,
                              void* d_out, int out_size, void* d_ws, size_t ws_size,
                              hipStream_t stream) {
    const float* X   = (const float*)d_in[0];   // (16,3,256,256)
    const float* ew  = (const float*)d_in[1];   // (8,8,3,3,3)
    const float* eb  = (const float*)d_in[2];   // (8,8)
    const float* gw  = (const float*)d_in[3];   // (8,23)
    const float* gb  = (const float*)d_in[4];   // (8,)
    const float* lw  = (const float*)d_in[5];   // (512,1000)
    const float* lb  = (const float*)d_in[6];   // (1000,)
    float* out = (float*)d_out;                 // 16000 logits + 1 aux

    char* ws = (char*)d_ws;
    int*   e_idx  = (int*)  (ws + WS_EIDX);
    float* gate_p = (float*)(ws + WS_GATE);
    float* probs  = (float*)(ws + WS_PROBS);
    float* tokSum = (float*)(ws + WS_TOKSUM);
    float* feat   = (float*)(ws + WS_POOLED);

    k1_gate     <<<NTOK / 256, 256, 0, stream>>>(X, gw, gb, e_idx, gate_p, probs);
    k2_capacity <<<1,          256, 0, stream>>>(e_idx, gate_p, probs, out);
    k3_conv     <<<NTOK,       256, 0, stream>>>(X, ew, eb, e_idx, gate_p, tokSum);
    k4_pool     <<<8192 / 256, 256, 0, stream>>>(tokSum, feat);
    k5_gemm_wmma<<<(NCLS + 15) / 16, 32, 0, stream>>>(feat, lw, lb, out);
}